// TokenEmbedding_36713380446420
// MI455X (gfx1250) — compile-verified
//
#include <hip/hip_runtime.h>

typedef __attribute__((ext_vector_type(2))) float v2f;
typedef __attribute__((ext_vector_type(8))) float v8f;

#define TAO     3
#define MM      5
#define C_IN    7
#define D_MODEL 512
#define NKERN   73              // D_MODEL / C_IN
#define S_LEN   4096            // power of two -> cheap circular wrap
#define B_SZ    32
#define KTOT    18              // (MM+1) * 3 real taps
#define KPAD    20              // K padded to 5 x wmma 16x16x4 (col 18 = bias)
#define NPAD    80              // 73 (+1 w_left) kernels padded to 5 N-tiles
#define NTILES_T (S_LEN / 16)   // 256

__global__ __launch_bounds__(256)
void TokenEmbedding_wmma_kernel(const float* __restrict__ x,
                                const float* __restrict__ w_conv,
                                const float* __restrict__ b_conv,
                                const float* __restrict__ w_left,
                                const float* __restrict__ b_left,
                                float* __restrict__ out)
{
    // Padded weights in LDS: rows 0..72 w_conv, row 73 w_left,
    // col 18 = bias (paired with constant 1.0 in A), col 19 = 0.
    __shared__ float Wpad[NPAD * KPAD];          // 6.4 KB
    for (int i = threadIdx.x; i < NPAD * KPAD; i += 256) {
        const int o = i / KPAD;
        const int j = i - o * KPAD;
        float v = 0.0f;
        if (j < KTOT) {
            if (o < NKERN)       v = w_conv[o * KTOT + j];
            else if (o == NKERN) v = w_left[j];
        } else if (j == KTOT) {
            if (o < NKERN)       v = b_conv[o];
            else if (o == NKERN) v = b_left[0];
        }
        Wpad[i] = v;
    }
    __syncthreads();

    const int lane = threadIdx.x & 31;
    // Wave-uniform coordinates -> force into SGPRs for SADDR addressing.
    const int w_s = __builtin_amdgcn_readfirstlane(
        (int)(blockIdx.x * 8) + (int)(threadIdx.x >> 5));
    const int tt = w_s & (NTILES_T - 1);
    const int c  = (w_s >> 8) % C_IN;
    const int b  = w_s / (NTILES_T * C_IN);
    const int t0 = tt * 16;

    const int  row = lane & 15;
    const bool hi  = lane >= 16;                 // K-pair select (j0 vs j0+2)
    const int  t   = t0 + row;
    const int  tb  = t * C_IN;                   // element offset of row t

    const float* __restrict__ xb = x + (size_t)b * S_LEN * C_IN + c;

    // A operand (16x4 f32): row M = lane&15; lanes 0-15 K pair (j0,j0+1),
    // lanes 16-31 K pair (j0+2,j0+3). Tap offset for K index j = m*3+k is
    // d(j) = (j%3) - 1 - 3*(j/3), a compile-time constant per half-wave.
    v2f A[5];
    if (tt != 0 && tt != NTILES_T - 1) {
        // Fast path (254/256 tiles): t>=16 => no wrap, mask always passes,
        // idx = t + d(j) >= 0. One cndmask(const) + add + load per element.
#pragma unroll
        for (int kk = 0; kk < 5; ++kk) {
#pragma unroll
            for (int e = 0; e < 2; ++e) {
                const int jl = kk * 4 + e, jh = jl + 2;
                const int dl = (jl % 3) - 1 - 3 * (jl / 3);
                const int dh = (jh % 3) - 1 - 3 * (jh / 3);
                float val;
                if (jh < KTOT) {
                    const int doff7 = hi ? dh * C_IN : dl * C_IN;
                    val = xb[tb + doff7];
                } else {
                    // kk==4: low lanes load j=16,17; high lanes are the
                    // bias-one column (j=18 -> 1.0) / zero pad (j=19).
                    const float lv = xb[tb + dl * C_IN];
                    const float cv = (jh == KTOT) ? 1.0f : 0.0f;
                    val = hi ? cv : lv;
                }
                if (e == 0) A[kk].x = val; else A[kk].y = val;
            }
        }
    } else {
        // Edge tiles (tt==0, 255): circular wrap + front zero-mask.
#pragma unroll
        for (int kk = 0; kk < 5; ++kk) {
#pragma unroll
            for (int e = 0; e < 2; ++e) {
                const int jl = kk * 4 + e, jh = jl + 2;
                const int ku = hi ? (jh % 3) - 1 : (jl % 3) - 1;
                const int m3 = hi ? 3 * (jh / 3) : 3 * (jl / 3);
                const int u  = (t + ku + S_LEN) & (S_LEN - 1);
                int idx = u - m3; if (idx < 0) idx = 0;       // safe clamp
                const float lv = xb[idx * C_IN];              // unconditional
                const bool jv   = hi ? (jh < KTOT) : (jl < KTOT);
                const bool real = (u >= MM * TAO) && jv;
                const float cv  = (hi ? (jh == KTOT) : (jl == KTOT)) ? 1.0f : 0.0f;
                const float val = real ? lv : cv;
                if (e == 0) A[kk].x = val; else A[kk].y = val;
            }
        }
    }

    // B operand (4x16 f32) from LDS: col N = lane&15, K pair rows by half.
    const int n     = lane & 15;
    const int bbase = n * KPAD + ((lane >> 4) << 1);

    // Scalar store base + one per-lane offset; all else immediate offsets.
    // d = c*73 + ntile*16 + n; (o==73, c==6) lands at d=511 via same formula.
    float* __restrict__ outs =
        out + ((size_t)b * S_LEN + t0) * D_MODEL + c * NKERN;
    const int voff = (hi ? 8 * D_MODEL : 0) + n;

#pragma unroll
    for (int ntile = 0; ntile < 5; ++ntile) {
        v8f acc = {};
#pragma unroll
        for (int kk = 0; kk < 5; ++kk) {
            const v2f Bv = *reinterpret_cast<const v2f*>(
                &Wpad[bbase + ntile * 16 * KPAD + kk * 4]);   // ds_load_b64
            acc = __builtin_amdgcn_wmma_f32_16x16x4_f32(
                false, A[kk], false, Bv, (short)0, acc, false, false);
        }
        if (ntile < 4) {
            // o = ntile*16 + n <= 63 < 73: unconditional stores
#pragma unroll
            for (int g = 0; g < 8; ++g)
                outs[voff + ntile * 16 + g * D_MODEL] = acc[g];
        } else {
            const int o = 64 + n;
            const bool valid = (o < NKERN) || ((o == NKERN) && (c == C_IN - 1));
            if (valid) {
#pragma unroll
                for (int g = 0; g < 8; ++g)
                    outs[voff + 64 + g * D_MODEL] = acc[g];
            }
        }
    }
}

extern "C" void kernel_launch(void* const* d_in, const int* in_sizes, int n_in,
                              void* d_out, int out_size, void* d_ws, size_t ws_size,
                              hipStream_t stream) {
    (void)in_sizes; (void)n_in; (void)d_ws; (void)ws_size; (void)out_size;
    const float* x      = (const float*)d_in[0];
    const float* w_conv = (const float*)d_in[1];
    const float* b_conv = (const float*)d_in[2];
    const float* w_left = (const float*)d_in[3];
    const float* b_left = (const float*)d_in[4];
    float* out = (float*)d_out;

    // 32 * 7 * 256 = 57344 waves, 8 waves (256 threads) per block
    const int total_waves = B_SZ * C_IN * NTILES_T;
    const int blocks = total_waves / 8;          // 7168
    TokenEmbedding_wmma_kernel<<<blocks, 256, 0, stream>>>(
        x, w_conv, b_conv, w_left, b_left, out);
}